// GaranAttention_38233798869800
// MI455X (gfx1250) — compile-verified
//
#include <hip/hip_runtime.h>
#include <hip/hip_bf16.h>
#include <math.h>

typedef __attribute__((ext_vector_type(2))) float v2f;
typedef __attribute__((ext_vector_type(8))) float v8f;

#define DQ 512
#define DO 256
#define NH 8
#define DH 32
#define HW 1024
#define NB 64

#if defined(__HIP_DEVICE_COMPILE__) && \
    __has_builtin(__builtin_amdgcn_global_load_async_to_lds_b128) && \
    __has_builtin(__builtin_amdgcn_s_wait_asynccnt)
#define USE_ASYNC_LDS 1
#else
#define USE_ASYNC_LDS 0
#endif

#if USE_ASYNC_LDS
typedef int v4i_ __attribute__((vector_size(16)));
typedef __attribute__((address_space(1))) v4i_ GASv4;   // global int4*
typedef __attribute__((address_space(3))) v4i_ LASv4;   // LDS int4*
#endif

// ---------------------------------------------------------------------------
// Phase 1: qv = q @ w_q^T + b_q ; fold query through conv weights:
//   qkA[b][r][c] (r<8: w_kc heads, r>=8: w_kd heads), qb[b][r] = qv_head.bias
// ---------------------------------------------------------------------------
__global__ __launch_bounds__(256) void k_proj(
    const float* __restrict__ q, const float* __restrict__ w_q,
    const float* __restrict__ b_q,
    const float* __restrict__ w_kc, const float* __restrict__ b_kc,
    const float* __restrict__ w_kd, const float* __restrict__ b_kd,
    float* __restrict__ qkA, float* __restrict__ qb)
{
    __shared__ float qlds[DQ];
    __shared__ float qvl[DO];
    const int b = blockIdx.x, t = threadIdx.x;

    qlds[t]       = q[b * DQ + t];
    qlds[t + 256] = q[b * DQ + 256 + t];
    __syncthreads();

    float acc = b_q[t];
    const float4* wr4 = (const float4*)(w_q + (size_t)t * DQ);
    #pragma unroll 4
    for (int k = 0; k < DQ / 4; ++k) {
        const float4 wv = wr4[k];
        acc += wv.x * qlds[4 * k] + wv.y * qlds[4 * k + 1] +
               wv.z * qlds[4 * k + 2] + wv.w * qlds[4 * k + 3];
    }
    qvl[t] = acc;
    __syncthreads();

    for (int r = 0; r < 16; ++r) {
        const float* W = (r < 8) ? w_kc : w_kd;
        const int head = r & 7;
        float s = 0.f;
        #pragma unroll
        for (int d = 0; d < DH; ++d)
            s += qvl[head * DH + d] * W[(size_t)(head * DH + d) * DO + t];
        qkA[((size_t)b * 16 + r) * DO + t] = s;
    }
    if (t < 16) {
        const float* bias = (t < 8) ? b_kc : b_kd;
        const int head = t & 7;
        float s = 0.f;
        #pragma unroll
        for (int d = 0; d < DH; ++d) s += qvl[head * DH + d] * bias[head * DH + d];
        qb[b * 16 + t] = s;
    }
}

// ---------------------------------------------------------------------------
// Phase 2: logits GEMM (16 x 256) @ (256 x 1024) with f32 WMMA.
//   grid (8 s-groups, 64 batches), 256 threads = 8 waves, 1 s-tile per wave.
//   v streamed through double-buffered LDS chunks (16 k-rows x 128 cols),
//   loaded coalesced b128 (async-to-LDS when the toolchain supports it).
// ---------------------------------------------------------------------------
#define APAD 260   // A row stride: lanes spread 4*l16 banks -> conflict-free
#define BSTR 136   // B chunk row stride: half-waves 2 rows apart -> +16 banks

__global__ __launch_bounds__(256) void k_logits(
    const float* __restrict__ V,          // (64, 256, 1024)
    const float* __restrict__ qkA, const float* __restrict__ qb,
    const float* __restrict__ tc, const float* __restrict__ td,
    float* __restrict__ lgw)              // (64, 16, 1024) scaled logits
{
    __shared__ float A[16 * APAD];
    __shared__ float BT[2][16 * BSTR];
    __shared__ float QB[16];

    const int sq = blockIdx.x;            // 128-column s-group
    const int b  = blockIdx.y;
    const int t = threadIdx.x;
    const int wave = t >> 5, lane = t & 31;
    const int half = lane >> 4, l16 = lane & 15;

    for (int i = t; i < 16 * DO; i += 256) {
        const int r = i >> 8, c = i & 255;
        A[r * APAD + c] = qkA[((size_t)b * 16 + r) * DO + c];
    }
    if (t < 16) QB[t] = qb[b * 16 + t];

    const float* Vb = V + (size_t)b * DO * HW + sq * 128;
    const int kr0  = t >> 5;              // slice 0 row   (slice = t)
    const int kr1  = (t + 256) >> 5;      // slice 1 row   (slice = t + 256)
    const int col  = (t & 31) * 4;

    // stage one 16x128 chunk (rows k0..k0+15) into buffer `buf`
    auto stage = [&](int k0, int buf) {
        const float* s0 = Vb + (size_t)(k0 + kr0) * HW + col;
        const float* s1 = Vb + (size_t)(k0 + kr1) * HW + col;
        float* d0 = &BT[buf][kr0 * BSTR + col];
        float* d1 = &BT[buf][kr1 * BSTR + col];
#if USE_ASYNC_LDS
        __builtin_amdgcn_global_load_async_to_lds_b128((GASv4*)s0, (LASv4*)d0, 0, 0);
        __builtin_amdgcn_global_load_async_to_lds_b128((GASv4*)s1, (LASv4*)d1, 0, 0);
#else
        *(float4*)d0 = *(const float4*)s0;
        *(float4*)d1 = *(const float4*)s1;
#endif
    };

    v8f acc = {};
    const int sw = wave * 16;

    stage(0, 0);
    for (int c = 0; c < 16; ++c) {
        const int k0 = c * 16;
        if (c < 15) {
            stage(k0 + 16, (c + 1) & 1);
#if USE_ASYNC_LDS
            __builtin_amdgcn_s_wait_asynccnt(2);   // current chunk landed
#endif
        } else {
#if USE_ASYNC_LDS
            __builtin_amdgcn_s_wait_asynccnt(0);
#endif
        }
        __syncthreads();
        const float* B0 = BT[c & 1];
        #pragma unroll
        for (int ks = 0; ks < 4; ++ks) {
            const int kb = ks * 4 + half * 2;
            v2f a, bb;
            a.x  = A[l16 * APAD + k0 + kb];
            a.y  = A[l16 * APAD + k0 + kb + 1];
            bb.x = B0[kb * BSTR + sw + l16];
            bb.y = B0[(kb + 1) * BSTR + sw + l16];
            acc = __builtin_amdgcn_wmma_f32_16x16x4_f32(
                false, a, false, bb, (short)0, acc, false, false);
        }
        __syncthreads();   // protect buffer reuse before next stage()
    }

    const float itc = 1.f / tc[0];
    const float itd = 1.f / td[0];
    // C layout: VGPR r -> row r (lanes 0-15) / row r+8 (lanes 16-31)
    #pragma unroll
    for (int r = 0; r < 8; ++r) {
        const int row = r + 8 * half;
        const float it = (row < 8) ? itc : itd;
        lgw[((size_t)b * 16 + row) * HW + sq * 128 + sw + l16] =
            (acc[r] + QB[row]) * it;
    }
}

// ---------------------------------------------------------------------------
// Phase 2b: per-row softmax (rows 0..7 -> attn) / sigmoid (rows 8..15 -> dif)
//   grid (16 rows, 64 batches), 256 threads, float4 I/O.
// ---------------------------------------------------------------------------
__global__ __launch_bounds__(256) void k_sm(
    const float* __restrict__ lgw, float* __restrict__ attn,
    float* __restrict__ dif)
{
    __shared__ float red[8];
    __shared__ float bc;
    const int row = blockIdx.x, b = blockIdx.y;
    const int t = threadIdx.x, wave = t >> 5, lane = t & 31;

    float4 x = ((const float4*)(lgw + ((size_t)b * 16 + row) * HW))[t];

    if (row < 8) {
        float m = fmaxf(fmaxf(x.x, x.y), fmaxf(x.z, x.w));
        #pragma unroll
        for (int o = 16; o > 0; o >>= 1) m = fmaxf(m, __shfl_xor(m, o, 32));
        if (lane == 0) red[wave] = m;
        __syncthreads();
        if (t == 0) {
            float mm = red[0];
            #pragma unroll
            for (int i = 1; i < 8; ++i) mm = fmaxf(mm, red[i]);
            bc = mm;
        }
        __syncthreads();
        m = bc;
        x.x = __expf(x.x - m); x.y = __expf(x.y - m);
        x.z = __expf(x.z - m); x.w = __expf(x.w - m);
        float s = (x.x + x.y) + (x.z + x.w);
        #pragma unroll
        for (int o = 16; o > 0; o >>= 1) s += __shfl_xor(s, o, 32);
        if (lane == 0) red[wave] = s;
        __syncthreads();
        if (t == 0) {
            float ss = 0.f;
            #pragma unroll
            for (int i = 0; i < 8; ++i) ss += red[i];
            bc = 1.f / ss;
        }
        __syncthreads();
        const float inv = bc;
        x.x *= inv; x.y *= inv; x.z *= inv; x.w *= inv;
        ((float4*)(attn + ((size_t)b * NH + row) * HW))[t] = x;
    } else {
        x.x = 1.f / (1.f + __expf(-x.x));
        x.y = 1.f / (1.f + __expf(-x.y));
        x.z = 1.f / (1.f + __expf(-x.z));
        x.w = 1.f / (1.f + __expf(-x.w));
        ((float4*)(dif + ((size_t)b * NH + row - 8) * HW))[t] = x;
    }
}

// ---------------------------------------------------------------------------
// Phase 3: av[b][n][c] = sum_s attn[b][n][s] * v[b][c][s]   (float4 streams)
// ---------------------------------------------------------------------------
__global__ __launch_bounds__(256) void k_av(
    const float* __restrict__ V, const float* __restrict__ attn,
    float* __restrict__ av)
{
    __shared__ float AT[NH * HW];
    const int cg = blockIdx.x, b = blockIdx.y;
    const int t = threadIdx.x, wave = t >> 5, lane = t & 31;

    for (int i = t; i < NH * HW / 4; i += 256)
        ((float4*)AT)[i] = ((const float4*)(attn + (size_t)b * NH * HW))[i];
    __syncthreads();

    const float4* AT4 = (const float4*)AT;
    for (int rr = 0; rr < 4; ++rr) {
        const int c = cg * 32 + wave * 4 + rr;
        const float4* Vr4 = (const float4*)(V + ((size_t)b * DO + c) * HW);
        float acc[NH] = {};
        for (int j = 0; j < 8; ++j) {
            const float4 sv = Vr4[j * 32 + lane];
            #pragma unroll
            for (int n = 0; n < NH; ++n) {
                const float4 at = AT4[n * 256 + j * 32 + lane];
                acc[n] += sv.x * at.x + sv.y * at.y + sv.z * at.z + sv.w * at.w;
            }
        }
        #pragma unroll
        for (int n = 0; n < NH; ++n) {
            float s = acc[n];
            #pragma unroll
            for (int o = 16; o > 0; o >>= 1) s += __shfl_xor(s, o, 32);
            if (lane == 0) av[((size_t)b * NH + n) * DO + c] = s;
        }
    }
}

// ---------------------------------------------------------------------------
// Phase 3b: attnf[b][n*32+d] = b_v + sum_c w_v[n*32+d][c] * av[b][n][c]
// ---------------------------------------------------------------------------
__global__ __launch_bounds__(256) void k_attnf(
    const float* __restrict__ w_v, const float* __restrict__ b_v,
    const float* __restrict__ av, float* __restrict__ attnf)
{
    __shared__ float AV[NH * DO];
    const int b = blockIdx.x, t = threadIdx.x;
    for (int i = t; i < NH * DO / 4; i += 256)
        ((float4*)AV)[i] = ((const float4*)(av + (size_t)b * NH * DO))[i];
    __syncthreads();
    const int n = t >> 5;
    float acc = b_v[t];
    const float4* wr4 = (const float4*)(w_v + (size_t)t * DO);
    const float4* av4 = (const float4*)(AV + n * DO);
    #pragma unroll 4
    for (int c = 0; c < DO / 4; ++c) {
        const float4 w = wr4[c], a = av4[c];
        acc += w.x * a.x + w.y * a.y + w.z * a.z + w.w * a.w;
    }
    attnf[b * DO + t] = acc;
}

// ---------------------------------------------------------------------------
// Phase 4: out[b,ch,p,q] = leaky(bn( dif[b,n,q*32+p]*attnf[b,ch] + v[b,ch,p,q] ))
//   float4 in/out; dif transposed through 33-stride padded LDS.
// ---------------------------------------------------------------------------
__global__ __launch_bounds__(256) void k_out(
    const float* __restrict__ V, const float* __restrict__ dif,
    const float* __restrict__ attnf,
    const float* __restrict__ gamma, const float* __restrict__ beta,
    const float* __restrict__ mean, const float* __restrict__ var,
    float* __restrict__ out)
{
    __shared__ float TL[33 * 32];
    const int ch = blockIdx.x, b = blockIdx.y;
    const int t = threadIdx.x;
    const int n = ch >> 5;

    // TL[33*y + x] = dif[x*32 + y]
    {
        const float4 d = ((const float4*)(dif + ((size_t)b * NH + n) * HW))[t];
        const int s0 = t * 4;
        TL[33 * ((s0 + 0) & 31) + ((s0 + 0) >> 5)] = d.x;
        TL[33 * ((s0 + 1) & 31) + ((s0 + 1) >> 5)] = d.y;
        TL[33 * ((s0 + 2) & 31) + ((s0 + 2) >> 5)] = d.z;
        TL[33 * ((s0 + 3) & 31) + ((s0 + 3) >> 5)] = d.w;
    }
    __syncthreads();

    const float a   = attnf[b * DO + ch];
    const float inv = gamma[ch] * rsqrtf(var[ch] + 1e-5f);
    const float mb  = mean[ch], bt = beta[ch];

    const int idx = t * 4;
    const int y = idx >> 5, x = idx & 31;   // x in {0,4,...,28}
    const float4 vv = ((const float4*)(V + ((size_t)b * DO + ch) * HW))[t];

    float4 r;
    r.x = TL[33 * y + x + 0] * a + vv.x;
    r.y = TL[33 * y + x + 1] * a + vv.y;
    r.z = TL[33 * y + x + 2] * a + vv.z;
    r.w = TL[33 * y + x + 3] * a + vv.w;
    r.x = (r.x - mb) * inv + bt;  r.x = (r.x >= 0.f) ? r.x : 0.1f * r.x;
    r.y = (r.y - mb) * inv + bt;  r.y = (r.y >= 0.f) ? r.y : 0.1f * r.y;
    r.z = (r.z - mb) * inv + bt;  r.z = (r.z >= 0.f) ? r.z : 0.1f * r.z;
    r.w = (r.w - mb) * inv + bt;  r.w = (r.w >= 0.f) ? r.w : 0.1f * r.w;
    ((float4*)(out + ((size_t)b * DO + ch) * HW))[t] = r;
}

// ---------------------------------------------------------------------------
extern "C" void kernel_launch(void* const* d_in, const int* in_sizes, int n_in,
                              void* d_out, int out_size, void* d_ws, size_t ws_size,
                              hipStream_t stream)
{
    const float* q     = (const float*)d_in[0];
    const float* v     = (const float*)d_in[1];
    const float* w_q   = (const float*)d_in[2];
    const float* b_q   = (const float*)d_in[3];
    const float* w_kc  = (const float*)d_in[4];
    const float* b_kc  = (const float*)d_in[5];
    const float* w_kd  = (const float*)d_in[6];
    const float* b_kd  = (const float*)d_in[7];
    const float* w_v   = (const float*)d_in[8];
    const float* b_v   = (const float*)d_in[9];
    const float* tc    = (const float*)d_in[10];
    const float* td    = (const float*)d_in[11];
    const float* gamma = (const float*)d_in[12];
    const float* beta  = (const float*)d_in[13];
    const float* mean  = (const float*)d_in[14];
    const float* var   = (const float*)d_in[15];
    float* out = (float*)d_out;

    // workspace layout (floats): ~10 MB
    float* ws    = (float*)d_ws;
    float* qkA   = ws;                        // 64*16*256
    float* qb    = qkA   + NB * 16 * DO;      // 64*16
    float* lgw   = qb    + NB * 16;           // 64*16*1024
    float* attn  = lgw   + NB * 16 * HW;      // 64*8*1024
    float* dif   = attn  + NB * NH * HW;      // 64*8*1024
    float* av    = dif   + NB * NH * HW;      // 64*8*256
    float* attnf = av    + NB * NH * DO;      // 64*256

    k_proj  <<<NB, 256, 0, stream>>>(q, w_q, b_q, w_kc, b_kc, w_kd, b_kd, qkA, qb);
    k_logits<<<dim3(8, NB), 256, 0, stream>>>(v, qkA, qb, tc, td, lgw);
    k_sm    <<<dim3(16, NB), 256, 0, stream>>>(lgw, attn, dif);
    k_av    <<<dim3(8, NB), 256, 0, stream>>>(v, attn, av);
    k_attnf <<<NB, 256, 0, stream>>>(w_v, b_v, av, attnf);
    k_out   <<<dim3(DO, NB), 256, 0, stream>>>(v, dif, attnf, gamma, beta, mean, var, out);
}